// Multi_populations_GPFA_41094247088591
// MI455X (gfx1250) — compile-verified
//
#include <hip/hip_runtime.h>

#define F_ 8
#define T_ 512
#define N_ 512
#define R_ 128
#define FT 4096
#define NB 64
#define NBLK (FT / NB)   // 64 block steps
#define LDT 68           // padded LDS stride for 64-wide tiles
#define LDB 132          // padded LDS stride for 128-wide RHS tiles
#define JIT 1e-3f

typedef __attribute__((ext_vector_type(2))) float v2f;
typedef __attribute__((ext_vector_type(8))) float v8f;

// LDS generic pointers carry the LDS byte offset in addr[31:0] (ISA 10.2).
__device__ __forceinline__ unsigned lds_addr(const void* p) {
    return (unsigned)(size_t)p;
}
// Async direct global->LDS copy (16B per lane), tracked by ASYNCcnt.
__device__ __forceinline__ void async_copy_b128(unsigned lds, const float* gaddr) {
    asm volatile("global_load_async_to_lds_b128 %0, %1, off" ::"v"(lds), "v"(gaddr)
                 : "memory");
}
__device__ __forceinline__ void wait_async0() {
    asm volatile("s_wait_asynccnt 0x0" ::: "memory");
}

// ---------------------------------------------------------------- build K_f
__global__ void build_K_kernel(const float* __restrict__ gamma, float* __restrict__ K) {
    int idx = blockIdx.x * blockDim.x + threadIdx.x;
    if (idx >= F_ * T_ * T_) return;
    int f = idx >> 18;            // T*T = 2^18
    int rem = idx & (T_ * T_ - 1);
    int t = rem >> 9, u = rem & (T_ - 1);
    float d = (float)(t - u);
    float v = (1.0f - JIT) * __expf(-0.5f * gamma[f] * d * d);
    if (t == u) v += JIT;
    K[idx] = v;
}

// ------------------------------------------------- c_rinv = C'R^-1, c_rinv_c
__global__ void crinv_crc_kernel(const float* __restrict__ Cm, const float* __restrict__ rdiag,
                                 float* __restrict__ crinv, float* __restrict__ crc) {
    int n = threadIdx.x;  // 512 threads
    float rv = 1.0f / rdiag[n];
#pragma unroll
    for (int f = 0; f < F_; ++f) crinv[f * N_ + n] = Cm[n * F_ + f] * rv;
    __syncthreads();
    if (n < F_ * F_) {
        int i = n >> 3, j = n & 7;
        float s = 0.f;
        for (int nn = 0; nn < N_; ++nn) s += crinv[i * N_ + nn] * Cm[nn * F_ + j];
        crc[n] = s;
    }
}

// ------------------------------------- in-place Gauss-Jordan per factor (SPD)
__global__ __launch_bounds__(1024) void gj_invert_kernel(float* __restrict__ K) {
    float* A = K + (size_t)blockIdx.x * T_ * T_;
    __shared__ float rowk[T_], colk[T_];
    __shared__ float pinv_s;
    const int tid = threadIdx.x, nt = blockDim.x;
    for (int k = 0; k < T_; ++k) {
        for (int j = tid; j < T_; j += nt) { rowk[j] = A[k * T_ + j]; colk[j] = A[j * T_ + k]; }
        __syncthreads();
        if (tid == 0) pinv_s = 1.0f / rowk[k];
        __syncthreads();
        float pinv = pinv_s;
        for (int e = tid; e < T_ * T_; e += nt) {
            int i = e >> 9, j = e & (T_ - 1);
            if (i != k && j != k) A[e] -= colk[i] * rowk[j] * pinv;
        }
        for (int j = tid; j < T_; j += nt) {
            A[k * T_ + j] = (j == k) ? pinv : rowk[j] * pinv;
            if (j != k) A[j * T_ + k] = -colk[j] * pinv;
        }
        __syncthreads();
    }
}

// -------- M[(t*F+i),(u*F+j)] = Kinv[i,t,u]*(i==j) + (t==u)*crc[i,j]  (64 MB)
__global__ void assemble_M_kernel(const float* __restrict__ Kinv, const float* __restrict__ crc,
                                  float* __restrict__ M) {
    size_t idx = (size_t)blockIdx.x * blockDim.x + threadIdx.x;
    if (idx >= (size_t)FT * FT) return;
    int row = (int)(idx >> 12);
    int col = (int)(idx & (FT - 1));
    int t = row >> 3, i = row & 7;
    int u = col >> 3, j = col & 7;
    float v = 0.f;
    if (i == j) v = Kinv[(size_t)i * T_ * T_ + (size_t)t * T_ + u];
    if (t == u) v += crc[i * F_ + j];
    M[idx] = v;
}

// ------------- term1[(t*F+f)*128 + r] = sum_n crinv[f,n]*(spike[r,n,t]-dm[n])
__global__ __launch_bounds__(256) void term1_kernel(const float* __restrict__ spike,
                                                    const float* __restrict__ dm,
                                                    const float* __restrict__ crinv,
                                                    float* __restrict__ B) {
    __shared__ float cs[F_ * N_];
    __shared__ float dms[N_];
    int tid = threadIdx.x;
    for (int e = tid; e < F_ * N_; e += 256) cs[e] = crinv[e];
    for (int e = tid; e < N_; e += 256) dms[e] = dm[e];
    __syncthreads();
    int r = blockIdx.x >> 1;
    int t = ((blockIdx.x & 1) << 8) + tid;
    float acc[F_];
#pragma unroll
    for (int f = 0; f < F_; ++f) acc[f] = 0.f;
    const float* sp = spike + (size_t)r * N_ * T_ + t;
    for (int n = 0; n < N_; ++n) {
        if (n + 8 < N_) __builtin_prefetch(sp + (size_t)(n + 8) * T_, 0, 0);
        float s = sp[(size_t)n * T_] - dms[n];
#pragma unroll
        for (int f = 0; f < F_; ++f) acc[f] += cs[f * N_ + n] * s;
    }
#pragma unroll
    for (int f = 0; f < F_; ++f) B[((size_t)(t * F_ + f)) * R_ + r] = acc[f];
}

// ---------------------------------------- Cholesky: diag block factor in LDS
__global__ __launch_bounds__(256) void potrf_diag_kernel(float* __restrict__ M, int kb) {
    __shared__ float A[NB * LDT];
    int tid = threadIdx.x;
    size_t base = ((size_t)kb * NB) * FT + (size_t)kb * NB;
    for (int e = tid; e < NB * NB; e += 256) {
        int r = e >> 6, c = e & 63;
        A[r * LDT + c] = M[base + (size_t)r * FT + c];
    }
    __syncthreads();
    for (int j = 0; j < NB; ++j) {
        if (tid == 0) A[j * LDT + j] = sqrtf(A[j * LDT + j]);
        __syncthreads();
        float dinv = 1.0f / A[j * LDT + j];
        for (int i = j + 1 + tid; i < NB; i += 256) A[i * LDT + j] *= dinv;
        __syncthreads();
        int w = NB - 1 - j;
        for (int e = tid; e < w * w; e += 256) {
            int i = j + 1 + e / w;
            int c = j + 1 + e % w;
            A[i * LDT + c] -= A[i * LDT + j] * A[c * LDT + j];
        }
        __syncthreads();
    }
    for (int e = tid; e < NB * NB; e += 256) {
        int r = e >> 6, c = e & 63;
        M[base + (size_t)r * FT + c] = A[r * LDT + c];
    }
}

// -------------------------------- panel: L21 = A21 * L11^-T (row recurrence)
__global__ __launch_bounds__(64) void trsm_panel_kernel(float* __restrict__ M, int kb) {
    __shared__ float Ls[NB * LDT];
    __shared__ float Xs[NB * LDT];
    int tid = threadIdx.x;  // 64
    int ib = kb + 1 + blockIdx.x;
    size_t dbase = ((size_t)kb * NB) * FT + (size_t)kb * NB;
    size_t pbase = ((size_t)ib * NB) * FT + (size_t)kb * NB;
    for (int e = tid; e < NB * NB; e += 64) {
        int r = e >> 6, c = e & 63;
        Ls[r * LDT + c] = M[dbase + (size_t)r * FT + c];
        Xs[r * LDT + c] = M[pbase + (size_t)r * FT + c];
    }
    __syncthreads();
    float* x = &Xs[tid * LDT];
    for (int j = 0; j < NB; ++j) {
        float s = x[j];
        for (int p = 0; p < j; ++p) s -= x[p] * Ls[j * LDT + p];
        x[j] = s / Ls[j * LDT + j];
    }
    __syncthreads();
    for (int e = tid; e < NB * NB; e += 64) {
        int r = e >> 6, c = e & 63;
        M[pbase + (size_t)r * FT + c] = Xs[r * LDT + c];
    }
}

// ----------------- trailing update A_ij -= L_ik * L_jk^T  (f32 WMMA 16x16x4)
__global__ __launch_bounds__(256) void syrk_trailing_kernel(float* __restrict__ M, int kb) {
    int ib = kb + 1 + blockIdx.x;
    int jb = kb + 1 + blockIdx.y;
    if (jb > ib) return;  // uniform per workgroup: EXEC stays all-ones for WMMA
    __shared__ float Ai[NB * LDT];
    __shared__ float Aj[NB * LDT];
    int tid = threadIdx.x;
    size_t ibase = ((size_t)ib * NB) * FT + (size_t)kb * NB;
    size_t jbase = ((size_t)jb * NB) * FT + (size_t)kb * NB;
    // direct global->LDS async staging (ASYNCcnt), 16B per lane
    for (int e = tid; e < (NB * NB) / 4; e += 256) {
        int r = e >> 4, c = (e & 15) << 2;
        async_copy_b128(lds_addr(&Ai[r * LDT + c]), &M[ibase + (size_t)r * FT + c]);
        async_copy_b128(lds_addr(&Aj[r * LDT + c]), &M[jbase + (size_t)r * FT + c]);
    }
    wait_async0();
    __syncthreads();
    int wave = tid >> 5, lane = tid & 31, hf = lane >> 4, ln = lane & 15;
    size_t obase = ((size_t)ib * NB) * FT + (size_t)jb * NB;
    for (int s = wave; s < 16; s += 8) {   // 4x4 grid of 16x16 subtiles
        int m0 = (s >> 2) << 4;
        int n0 = (s & 3) << 4;
        v8f acc = {};
#pragma unroll
        for (int kk = 0; kk < NB; kk += 4) {
            int k0 = kk + 2 * hf;  // A: VGPR0=K0/K2, VGPR1=K1/K3 (lanes 0-15 / 16-31)
            v2f a, b;
            a.x = Ai[(m0 + ln) * LDT + k0];
            a.y = Ai[(m0 + ln) * LDT + k0 + 1];
            b.x = Aj[(n0 + ln) * LDT + k0];      // B = L_jk^T, so read L_jk[n][k]
            b.y = Aj[(n0 + ln) * LDT + k0 + 1];
            acc = __builtin_amdgcn_wmma_f32_16x16x4_f32(false, a, false, b, (short)0, acc,
                                                        false, false);
        }
#pragma unroll
        for (int v = 0; v < 8; ++v) {  // C/D: VGPR v -> rows v (lanes 0-15) / v+8 (16-31)
            int row = m0 + v + 8 * hf;
            int col = n0 + ln;
            M[obase + (size_t)row * FT + col] -= acc[v];
        }
    }
}

// ------------- RHS update B_t -= op(L) * Y_k (WMMA); transA for backward sweep
__global__ __launch_bounds__(256) void gemm_update_kernel(const float* __restrict__ M,
                                                          float* __restrict__ B, int kb,
                                                          int transA) {
    __shared__ float As[NB * LDT];
    __shared__ float Bs[NB * LDB];
    int tid = threadIdx.x;
    int tb;
    size_t abase;
    if (!transA) {
        tb = kb + 1 + blockIdx.x;                      // forward: B_i -= L[i][kb] * Y_kb
        abase = ((size_t)tb * NB) * FT + (size_t)kb * NB;
    } else {
        tb = blockIdx.x;                               // backward: B_j -= L[kb][j]^T * X_kb
        abase = ((size_t)kb * NB) * FT + (size_t)tb * NB;
    }
    for (int e = tid; e < (NB * NB) / 4; e += 256) {
        int r = e >> 4, c = (e & 15) << 2;
        async_copy_b128(lds_addr(&As[r * LDT + c]), &M[abase + (size_t)r * FT + c]);
    }
    for (int e = tid; e < (NB * R_) / 4; e += 256) {
        int r = e >> 5, c = (e & 31) << 2;
        async_copy_b128(lds_addr(&Bs[r * LDB + c]), &B[((size_t)(kb * NB + r)) * R_ + c]);
    }
    wait_async0();
    __syncthreads();
    int wave = tid >> 5, lane = tid & 31, hf = lane >> 4, ln = lane & 15;
    for (int s = wave; s < 32; s += 8) {  // 4x8 grid of 16x16 subtiles (64x128 out)
        int m0 = (s >> 3) << 4;
        int n0 = (s & 7) << 4;
        v8f acc = {};
#pragma unroll
        for (int kk = 0; kk < NB; kk += 4) {
            int k0 = kk + 2 * hf;
            v2f a, b;
            if (!transA) {
                a.x = As[(m0 + ln) * LDT + k0];
                a.y = As[(m0 + ln) * LDT + k0 + 1];
            } else {
                a.x = As[k0 * LDT + m0 + ln];
                a.y = As[(k0 + 1) * LDT + m0 + ln];
            }
            b.x = Bs[k0 * LDB + n0 + ln];
            b.y = Bs[(k0 + 1) * LDB + n0 + ln];
            acc = __builtin_amdgcn_wmma_f32_16x16x4_f32(false, a, false, b, (short)0, acc,
                                                        false, false);
        }
#pragma unroll
        for (int v = 0; v < 8; ++v) {
            int row = tb * NB + m0 + v + 8 * hf;
            int col = n0 + ln;
            B[(size_t)row * R_ + col] -= acc[v];
        }
    }
}

// ------------------- diag triangular solve vs 128 RHS (fwd: L, bwd: L^T)
__global__ __launch_bounds__(128) void trsv_diag_kernel(const float* __restrict__ M,
                                                        float* __restrict__ B, int kb,
                                                        int backward) {
    __shared__ float Ls[NB * LDT];
    __shared__ float Ys[NB * R_];
    int tid = threadIdx.x;  // owns one RHS column
    size_t dbase = ((size_t)kb * NB) * FT + (size_t)kb * NB;
    for (int e = tid; e < NB * NB; e += 128) {
        int r = e >> 6, c = e & 63;
        Ls[r * LDT + c] = M[dbase + (size_t)r * FT + c];
    }
    for (int r = 0; r < NB; ++r) Ys[r * R_ + tid] = B[((size_t)(kb * NB + r)) * R_ + tid];
    __syncthreads();
    if (!backward) {
        for (int j = 0; j < NB; ++j) {
            float s = Ys[j * R_ + tid];
            for (int p = 0; p < j; ++p) s -= Ls[j * LDT + p] * Ys[p * R_ + tid];
            Ys[j * R_ + tid] = s / Ls[j * LDT + j];
        }
    } else {
        for (int j = NB - 1; j >= 0; --j) {
            float s = Ys[j * R_ + tid];
            for (int p = j + 1; p < NB; ++p) s -= Ls[p * LDT + j] * Ys[p * R_ + tid];
            Ys[j * R_ + tid] = s / Ls[j * LDT + j];
        }
    }
    for (int r = 0; r < NB; ++r) B[((size_t)(kb * NB + r)) * R_ + tid] = Ys[r * R_ + tid];
}

// ---------------------------------- out[r,f,t] = X[(t*F+f), r]
__global__ void writeout_kernel(const float* __restrict__ X, float* __restrict__ out) {
    int idx = blockIdx.x * blockDim.x + threadIdx.x;
    if (idx >= R_ * F_ * T_) return;
    int r = idx / (F_ * T_);
    int f = (idx / T_) & 7;
    int t = idx & (T_ - 1);
    out[idx] = X[((size_t)(t * F_ + f)) * R_ + r];
}

extern "C" void kernel_launch(void* const* d_in, const int* in_sizes, int n_in, void* d_out,
                              int out_size, void* d_ws, size_t ws_size, hipStream_t stream) {
    (void)in_sizes; (void)n_in; (void)out_size; (void)ws_size;
    const float* spike = (const float*)d_in[0];  // (128, 512, 512)
    const float* Cm    = (const float*)d_in[1];  // (512, 8)
    const float* dm    = (const float*)d_in[2];  // (512,)
    const float* rdiag = (const float*)d_in[3];  // (512,)
    const float* gamma = (const float*)d_in[4];  // (8,)
    float* out = (float*)d_out;

    // workspace partition (~78 MB total)
    float* K     = (float*)d_ws;                 // 8 * 512 * 512
    float* Mm    = K + (size_t)F_ * T_ * T_;     // 4096 * 4096
    float* crinv = Mm + (size_t)FT * FT;         // 8 * 512
    float* crc   = crinv + F_ * N_;              // 8 * 8
    float* Bm    = crc + F_ * F_;                // 4096 * 128 (term1 -> solution)

    build_K_kernel<<<(F_ * T_ * T_ + 255) / 256, 256, 0, stream>>>(gamma, K);
    crinv_crc_kernel<<<1, 512, 0, stream>>>(Cm, rdiag, crinv, crc);
    gj_invert_kernel<<<F_, 1024, 0, stream>>>(K);
    assemble_M_kernel<<<(int)(((size_t)FT * FT + 255) / 256), 256, 0, stream>>>(K, crc, Mm);
    term1_kernel<<<R_ * 2, 256, 0, stream>>>(spike, dm, crinv, Bm);

    // blocked right-looking Cholesky of M (in place, lower)
    for (int kb = 0; kb < NBLK; ++kb) {
        potrf_diag_kernel<<<1, 256, 0, stream>>>(Mm, kb);
        int rem = NBLK - 1 - kb;
        if (rem > 0) {
            trsm_panel_kernel<<<rem, 64, 0, stream>>>(Mm, kb);
            syrk_trailing_kernel<<<dim3(rem, rem), 256, 0, stream>>>(Mm, kb);
        }
    }
    // forward solve L Y = term1
    for (int kb = 0; kb < NBLK; ++kb) {
        trsv_diag_kernel<<<1, 128, 0, stream>>>(Mm, Bm, kb, 0);
        int rem = NBLK - 1 - kb;
        if (rem > 0) gemm_update_kernel<<<rem, 256, 0, stream>>>(Mm, Bm, kb, 0);
    }
    // backward solve L^T X = Y
    for (int kb = NBLK - 1; kb >= 0; --kb) {
        trsv_diag_kernel<<<1, 128, 0, stream>>>(Mm, Bm, kb, 1);
        if (kb > 0) gemm_update_kernel<<<kb, 256, 0, stream>>>(Mm, Bm, kb, 1);
    }
    writeout_kernel<<<(R_ * F_ * T_ + 255) / 256, 256, 0, stream>>>(Bm, out);
}